// AblationLayer_56358560858377
// MI455X (gfx1250) — compile-verified
//
#include <hip/hip_runtime.h>
#include <stdint.h>

#define THREADS 256
#define MAX_COPY_BLOCKS 8192

// ---- order-preserving float <-> unsigned mapping (for unsigned atomicMin) ----
__device__ __forceinline__ unsigned f2ord(float f) {
    unsigned u = __float_as_uint(f);
    return (u & 0x80000000u) ? ~u : (u | 0x80000000u);
}
__device__ __forceinline__ float ord2f(unsigned o) {
    return (o & 0x80000000u) ? __uint_as_float(o & 0x7fffffffu)
                             : __uint_as_float(~o);
}

// ---- kernel 0: reset the min accumulator (ws is poisoned / stale) ----
__global__ void ablation_init_ws(unsigned* ws) {
    if (threadIdx.x == 0 && blockIdx.x == 0) ws[0] = 0xFFFFFFFFu; // identity for unsigned min
}

// ---- kernel 1: fused streaming copy (CDNA5 async global<->LDS movers, NT hints) + global min ----
__global__ __launch_bounds__(THREADS)
void ablation_copy_min(const float4* __restrict__ x, float4* __restrict__ out,
                       size_t n4, unsigned* __restrict__ ws) {
    __shared__ float4   stage[2][THREADS];   // double-buffered staging tiles
    __shared__ unsigned wave_min[THREADS / 32];

    const unsigned tid = threadIdx.x;
    // Low 32 bits of the generic address of a __shared__ object are the LDS byte offset.
    const unsigned lds0 = (unsigned)(uintptr_t)(&stage[0][tid]);
    const unsigned lds1 = (unsigned)(uintptr_t)(&stage[1][tid]);

    float vmin = __uint_as_float(0x7f800000u); // +inf
    const size_t stride = (size_t)gridDim.x * THREADS;
    size_t i = (size_t)blockIdx.x * THREADS + tid;
    int buf = 0;

    for (; i < n4; i += stride, buf ^= 1) {
        const unsigned lds = buf ? lds1 : lds0;
        const unsigned long long src = (unsigned long long)(const void*)(x + i);
        const unsigned long long dst = (unsigned long long)(void*)(out + i);

        // async copy: global -> LDS (ASYNCcnt++), 16B/lane, non-temporal (zero reuse stream)
        asm volatile("global_load_async_to_lds_b128 %0, %1, off th:TH_LOAD_NT"
                     :: "v"(lds), "v"(src) : "memory");
        // drain: current load landed in LDS AND the previous iteration's async
        // store has finished reading its (other-buffer) LDS tile
        asm volatile("s_wait_asynccnt 0x0" ::: "memory");

        const float4 v = buf ? stage[1][tid] : stage[0][tid]; // ds_load_b128

        // async copy: LDS -> global (store leg never touches VGPRs), non-temporal
        asm volatile("global_store_async_from_lds_b128 %0, %1, off th:TH_STORE_NT"
                     :: "v"(dst), "v"(lds) : "memory");

        vmin = fminf(fminf(fminf(v.x, v.y), fminf(v.z, v.w)), vmin);
    }
    // drain outstanding async stores before exit
    asm volatile("s_wait_asynccnt 0x0" ::: "memory");

    // wave32 min tree
    for (int off = 16; off > 0; off >>= 1)
        vmin = fminf(vmin, __shfl_xor(vmin, off, 32));

    const int lane = tid & 31;
    const int wave = tid >> 5;
    if (lane == 0) wave_min[wave] = f2ord(vmin);
    __syncthreads();

    if (wave == 0) {
        unsigned o = (lane < (THREADS / 32)) ? wave_min[lane] : 0xFFFFFFFFu;
        for (int off = 4; off > 0; off >>= 1) {
            unsigned p = __shfl_xor(o, off, 32);
            o = (p < o) ? p : o;
        }
        if (lane == 0) atomicMin(ws, o); // global_atomic_min_u32
    }
}

// ---- kernel 2: scatter-fill the 32 ablated rows with m0 - (i+1)*1e7 ----
__global__ __launch_bounds__(THREADS)
void ablation_scatter(float* __restrict__ out, const int* __restrict__ idx,
                      const unsigned* __restrict__ ws, int C, int HW) {
    const int b = blockIdx.x;                       // batch index 0..B-1
    const float m0 = ord2f(ws[0]);
    const float val = (m0 == 0.0f) ? 0.0f : (m0 - (float)(b + 1) * 1.0e7f);
    const int c = idx[b];

    float* row = out + ((size_t)b * (size_t)C + (size_t)c) * (size_t)HW;
    const int hw4 = HW >> 2;
    float4* p4 = (float4*)row;
    const float4 v4 = make_float4(val, val, val, val);
    for (int t = threadIdx.x; t < hw4; t += blockDim.x) p4[t] = v4;
    for (int t = (hw4 << 2) + (int)threadIdx.x; t < HW; t += blockDim.x) row[t] = val;
}

extern "C" void kernel_launch(void* const* d_in, const int* in_sizes, int n_in,
                              void* d_out, int out_size, void* d_ws, size_t ws_size,
                              hipStream_t stream) {
    const float* x  = (const float*)d_in[0];
    const int* idx  = (const int*)d_in[1];   // indices (int32 per harness convention)
    float* out      = (float*)d_out;
    unsigned* ws    = (unsigned*)d_ws;

    const size_t n  = (size_t)out_size;      // 32*1024*56*56 = 102,760,448
    const size_t n4 = n >> 2;                // float4 count
    const int B  = in_sizes[1];              // 32
    const int C  = 1024;
    const int HW = (int)(n / ((size_t)B * (size_t)C)); // 3136

    size_t want = (n4 + THREADS - 1) / THREADS;
    const int copy_blocks = (want > MAX_COPY_BLOCKS) ? MAX_COPY_BLOCKS : (int)want;

    ablation_init_ws<<<1, 1, 0, stream>>>(ws);
    ablation_copy_min<<<copy_blocks, THREADS, 0, stream>>>(
        (const float4*)x, (float4*)out, n4, ws);
    ablation_scatter<<<B, THREADS, 0, stream>>>(out, idx, ws, C, HW);
}